// MambaBranch_43765716746546
// MI455X (gfx1250) — compile-verified
//
#include <hip/hip_runtime.h>
#include <hip/hip_bf16.h>
#include <math.h>

// ---------------------------------------------------------------------------
// VMamba SS2D forward for MI455X (gfx1250), fp32 end-to-end.
//   B=2, D_MODEL=96, D_INNER=192, D_STATE=16, DT_RANK=6, K=4, H=W=64, L=4096
// GEMMs use v_wmma_f32_16x16x4_f32; the selective scan uses a 3-phase
// chunked parallel scan (64 segments of 64 steps).  LDS staging in the scan
// kernels uses GLOBAL_LOAD_ASYNC_TO_LDS_B128 (ASYNCcnt) when available.
// ---------------------------------------------------------------------------

#define BSZ     2
#define DM      96
#define DI      192
#define NS      16
#define DTR     6
#define KDIR    4
#define HH      64
#define WW      64
#define LL      4096
#define NSEG    64
#define SEGL    64
#define CROWS   38          // DT_RANK + 2*D_STATE

typedef float v2f __attribute__((ext_vector_type(2)));
typedef float v8f __attribute__((ext_vector_type(8)));

#if defined(__has_builtin)
#if __has_builtin(__builtin_amdgcn_global_load_async_to_lds_b128) && \
    __has_builtin(__builtin_amdgcn_s_wait_asynccnt)
#define USE_ASYNC_LDS 1
#endif
#endif

#if defined(USE_ASYNC_LDS)
// Builtin signature (from compiler diagnostic): first param is
// 'int __attribute__((vector_size(16))) addrspace(1)*'; second is the LDS
// destination in addrspace(3).  Use GCC-vector typedefs so canonical types
// match, and route casts through uintptr_t (generic LDS addr low 32 bits are
// the LDS offset per the flat-aperture truncation rule).
typedef int b128_t __attribute__((vector_size(16)));
typedef __attribute__((address_space(1))) b128_t* gas_b128;
typedef __attribute__((address_space(3))) b128_t* las_b128;
#endif

// Stage nf4 float4-chunks of contiguous global data into LDS.
// Both pointers must be 16B aligned.  Caller must s_wait + barrier.
__device__ __forceinline__ void stage_lds(const float* __restrict__ g,
                                          float* __restrict__ l,
                                          int nf4, int tid)
{
#if defined(USE_ASYNC_LDS)
    for (int i = tid; i < nf4; i += DI)
        __builtin_amdgcn_global_load_async_to_lds_b128(
            (gas_b128)(uintptr_t)(g + 4 * i),
            (las_b128)(uintptr_t)(l + 4 * i), 0, 0);
#else
    for (int i = tid; i < nf4 * 4; i += DI) l[i] = g[i];
#endif
}

__device__ __forceinline__ void stage_wait()
{
#if defined(USE_ASYNC_LDS)
    __builtin_amdgcn_s_wait_asynccnt(0);
#endif
    __syncthreads();
}

// map scan-sequence index -> spatial row-major index for direction k
__device__ __forceinline__ int map_l(int k, int l) {
    int t = (k >= 2) ? (LL - 1 - l) : l;
    if (k & 1) t = ((t & 63) << 6) | (t >> 6);   // 64x64 transpose
    return t;
}

__device__ __forceinline__ float softplus_f(float x) {
    return (x > 20.f) ? x : log1pf(__expf(x));
}

// ---------------------------------------------------------------------------
// Kernel 1: in_proj (1x1 conv, 192x96) + BN  ->  x1 (B,192,L) row-major
// grid (B, L/64, 3 row-groups of 64), block 128 (4 waves, one l-tile each)
// ---------------------------------------------------------------------------
__global__ __launch_bounds__(128) void k_inproj(
    const float* __restrict__ x, const float* __restrict__ in_w,
    const float* __restrict__ bn_g, const float* __restrict__ bn_b,
    float* __restrict__ x1)
{
    const int lane = threadIdx.x & 31, wv = threadIdx.x >> 5;
    const int b = blockIdx.x, mg = blockIdx.z;
    const int l0 = blockIdx.y * 64 + wv * 16;
    const int n = lane & 15, hi = lane >> 4, kofs = hi * 2, ml = lane & 15;

    v8f acc[4] = {};
    const float* xb = x + (size_t)b * DM * LL + l0 + n;
    for (int k0 = 0; k0 < DM; k0 += 4) {
        v2f bv = { xb[(size_t)(k0 + kofs) * LL],
                   xb[(size_t)(k0 + kofs + 1) * LL] };
#pragma unroll
        for (int mt = 0; mt < 4; ++mt) {
            const int o = mg * 64 + mt * 16 + ml;
            v2f av = *(const v2f*)(in_w + (size_t)o * DM + k0 + kofs);
            acc[mt] = __builtin_amdgcn_wmma_f32_16x16x4_f32(
                false, av, false, bv, (short)0, acc[mt], false, false);
        }
    }
    const float inv = rsqrtf(1.f + 1e-5f);
#pragma unroll
    for (int mt = 0; mt < 4; ++mt)
#pragma unroll
        for (int r = 0; r < 8; ++r) {
            const int o = mg * 64 + mt * 16 + r + hi * 8;
            x1[(size_t)b * DI * LL + (size_t)o * LL + l0 + n] =
                acc[mt][r] * (bn_g[o] * inv) + bn_b[o];
        }
}

// ---------------------------------------------------------------------------
// Kernel 2: depthwise 3x3 + bias + SiLU -> u in (B, L, D) layout (transposed)
// ---------------------------------------------------------------------------
__global__ __launch_bounds__(256) void k_dwconv(
    const float* __restrict__ x1, const float* __restrict__ dww,
    const float* __restrict__ dwb, float* __restrict__ u)
{
    const int idx = blockIdx.x * 256 + threadIdx.x;
    if (idx >= BSZ * DI * LL) return;
    const int w  = idx & 63;
    const int h  = (idx >> 6) & 63;
    const int bd = idx >> 12;
    const int d  = bd % DI;
    const int b  = bd / DI;

    const float* base = x1 + (size_t)bd * LL;
    float acc = dwb[d];
#pragma unroll
    for (int kh = -1; kh <= 1; ++kh)
#pragma unroll
        for (int kw = -1; kw <= 1; ++kw) {
            const int hh = h + kh, ww2 = w + kw;
            if (hh >= 0 && hh < HH && ww2 >= 0 && ww2 < WW)
                acc += dww[d * 9 + (kh + 1) * 3 + (kw + 1)] * base[hh * WW + ww2];
        }
    const float s = acc * (1.f / (1.f + __expf(-acc)));   // SiLU
    u[((size_t)b * LL + h * WW + w) * DI + d] = s;
}

// ---------------------------------------------------------------------------
// Kernel 3: x_dbl GEMM (38 x 4096 per (b,k), K=192) via WMMA.
//   out rows 0..5 -> dtr (BK,L,6), 6..21 -> Bs (BK,L,16), 22..37 -> Cs
// grid (B*K, L/64), block 128
// ---------------------------------------------------------------------------
__global__ __launch_bounds__(128) void k_xdbl(
    const float* __restrict__ u, const float* __restrict__ xpw,
    float* __restrict__ dtr, float* __restrict__ Bsb, float* __restrict__ Csb)
{
    const int lane = threadIdx.x & 31, wv = threadIdx.x >> 5;
    const int bk = blockIdx.x, b = bk >> 2, k = bk & 3;
    const int l0 = blockIdx.y * 64 + wv * 16;
    const int n = lane & 15, hi = lane >> 4, kofs = hi * 2, ml = lane & 15;

    const int lsp = map_l(k, l0 + n);
    const float* up = u + ((size_t)b * LL + lsp) * DI + kofs;

    const float* ap[3];
    bool valid[3];
    v8f acc[3] = {};
    const v2f zv = { 0.f, 0.f };
#pragma unroll
    for (int mt = 0; mt < 3; ++mt) {
        const int c = mt * 16 + ml;
        valid[mt] = (c < CROWS);
        ap[mt] = xpw + ((size_t)k * CROWS + (valid[mt] ? c : 0)) * DI + kofs;
    }
    for (int k0 = 0; k0 < DI; k0 += 4) {
        v2f bv = *(const v2f*)(up + k0);
#pragma unroll
        for (int mt = 0; mt < 3; ++mt) {
            v2f av = valid[mt] ? *(const v2f*)(ap[mt] + k0) : zv;
            acc[mt] = __builtin_amdgcn_wmma_f32_16x16x4_f32(
                false, av, false, bv, (short)0, acc[mt], false, false);
        }
    }
    const size_t lbase = (size_t)bk * LL + l0 + n;
#pragma unroll
    for (int mt = 0; mt < 3; ++mt)
#pragma unroll
        for (int r = 0; r < 8; ++r) {
            const int c = mt * 16 + r + hi * 8;
            const float val = acc[mt][r];
            if (c < DTR)            dtr[lbase * DTR + c] = val;
            else if (c < DTR + NS)  Bsb[lbase * NS + (c - DTR)] = val;
            else if (c < CROWS)     Csb[lbase * NS + (c - DTR - NS)] = val;
        }
}

// ---------------------------------------------------------------------------
// Kernel 4: scan phase 1 — per segment compute transfer (P,Q) per (bk,d,n).
// grid (B*K, NSEG), block 192 (thread = channel d)
// ---------------------------------------------------------------------------
__global__ __launch_bounds__(192) void k_scan1(
    const float* __restrict__ u, const float* __restrict__ dtr,
    const float* __restrict__ Bsb, const float* __restrict__ dtw,
    const float* __restrict__ dtb, const float* __restrict__ A_logs,
    float* __restrict__ segPQ)
{
    const int bk = blockIdx.x, seg = blockIdx.y, d = threadIdx.x;
    const int b = bk >> 2, k = bk & 3;
    __shared__ __align__(16) float s_dtr[SEGL * DTR];
    __shared__ __align__(16) float s_B[SEGL * NS];
    const size_t lb = (size_t)bk * LL + seg * SEGL;
    // async DMA of contiguous chunks into LDS; overlapped with setup below
    stage_lds(dtr + lb * DTR, s_dtr, SEGL * DTR / 4, d);
    stage_lds(Bsb + lb * NS,  s_B,   SEGL * NS / 4,  d);

    const int kd = k * DI + d;
    float wr[DTR];
#pragma unroll
    for (int r = 0; r < DTR; ++r) wr[r] = dtw[(size_t)kd * DTR + r];
    const float bias = dtb[kd];
    float An[NS];
#pragma unroll
    for (int nn = 0; nn < NS; ++nn) An[nn] = -__expf(A_logs[(size_t)kd * NS + nn]);

    float P[NS], Q[NS];
#pragma unroll
    for (int nn = 0; nn < NS; ++nn) { P[nn] = 1.f; Q[nn] = 0.f; }
    stage_wait();

    const float* ub = u + (size_t)b * LL * DI + d;
    for (int t = 0; t < SEGL; ++t) {
        const int l = seg * SEGL + t;
        if (t + 8 < SEGL)
            __builtin_prefetch(ub + (size_t)map_l(k, l + 8) * DI, 0, 1);
        float dtv = bias;
#pragma unroll
        for (int r = 0; r < DTR; ++r) dtv += wr[r] * s_dtr[t * DTR + r];
        const float dt = softplus_f(dtv);
        const float uval = ub[(size_t)map_l(k, l) * DI];
        const float du = dt * uval;
#pragma unroll
        for (int nn = 0; nn < NS; ++nn) {
            const float e = __expf(dt * An[nn]);
            P[nn] *= e;
            Q[nn] = e * Q[nn] + du * s_B[t * NS + nn];
        }
    }
    const size_t ob = (((size_t)bk * NSEG + seg) * DI + d) * 32;
#pragma unroll
    for (int nn = 0; nn < NS; ++nn) {
        segPQ[ob + nn]      = P[nn];
        segPQ[ob + 16 + nn] = Q[nn];
    }
}

// ---------------------------------------------------------------------------
// Kernel 5: scan phase 2 — sequential combine over NSEG segments.
// Overwrites the P half of each slot with the segment's *incoming* state.
// grid (B*K), block 192
// ---------------------------------------------------------------------------
__global__ __launch_bounds__(192) void k_scanmid(float* __restrict__ segPQ)
{
    const int bk = blockIdx.x, d = threadIdx.x;
    float h[NS];
#pragma unroll
    for (int nn = 0; nn < NS; ++nn) h[nn] = 0.f;
    for (int s = 0; s < NSEG; ++s) {
        const size_t base = (((size_t)bk * NSEG + s) * DI + d) * 32;
#pragma unroll
        for (int nn = 0; nn < NS; ++nn) {
            const float Pv = segPQ[base + nn];
            const float Qv = segPQ[base + 16 + nn];
            segPQ[base + nn] = h[nn];          // store incoming state
            h[nn] = Pv * h[nn] + Qv;
        }
    }
}

// ---------------------------------------------------------------------------
// Kernel 6: scan phase 3 — replay segments with known h0, emit y (+ D*u).
// ys layout (BK, L, D).  grid (B*K, NSEG), block 192
// ---------------------------------------------------------------------------
__global__ __launch_bounds__(192) void k_scan2(
    const float* __restrict__ u, const float* __restrict__ dtr,
    const float* __restrict__ Bsb, const float* __restrict__ Csb,
    const float* __restrict__ dtw, const float* __restrict__ dtb,
    const float* __restrict__ A_logs, const float* __restrict__ Ds,
    const float* __restrict__ segPQ, float* __restrict__ ysb)
{
    const int bk = blockIdx.x, seg = blockIdx.y, d = threadIdx.x;
    const int b = bk >> 2, k = bk & 3;
    __shared__ __align__(16) float s_dtr[SEGL * DTR];
    __shared__ __align__(16) float s_B[SEGL * NS];
    __shared__ __align__(16) float s_C[SEGL * NS];
    const size_t lb = (size_t)bk * LL + seg * SEGL;
    stage_lds(dtr + lb * DTR, s_dtr, SEGL * DTR / 4, d);
    stage_lds(Bsb + lb * NS,  s_B,   SEGL * NS / 4,  d);
    stage_lds(Csb + lb * NS,  s_C,   SEGL * NS / 4,  d);

    const int kd = k * DI + d;
    float wr[DTR];
#pragma unroll
    for (int r = 0; r < DTR; ++r) wr[r] = dtw[(size_t)kd * DTR + r];
    const float bias = dtb[kd];
    const float Dv = Ds[kd];
    float An[NS];
#pragma unroll
    for (int nn = 0; nn < NS; ++nn) An[nn] = -__expf(A_logs[(size_t)kd * NS + nn]);

    float h[NS];
    const size_t hb = (((size_t)bk * NSEG + seg) * DI + d) * 32;
#pragma unroll
    for (int nn = 0; nn < NS; ++nn) h[nn] = segPQ[hb + nn];
    stage_wait();

    const float* ub = u + (size_t)b * LL * DI + d;
    float* yb = ysb + (size_t)bk * LL * DI + d;
    for (int t = 0; t < SEGL; ++t) {
        const int l = seg * SEGL + t;
        if (t + 8 < SEGL)
            __builtin_prefetch(ub + (size_t)map_l(k, l + 8) * DI, 0, 1);
        float dtv = bias;
#pragma unroll
        for (int r = 0; r < DTR; ++r) dtv += wr[r] * s_dtr[t * DTR + r];
        const float dt = softplus_f(dtv);
        const float uval = ub[(size_t)map_l(k, l) * DI];
        const float du = dt * uval;
        float y = 0.f;
#pragma unroll
        for (int nn = 0; nn < NS; ++nn) {
            const float e = __expf(dt * An[nn]);
            h[nn] = e * h[nn] + du * s_B[t * NS + nn];
            y += h[nn] * s_C[t * NS + nn];
        }
        yb[(size_t)l * DI] = y + Dv * uval;
    }
}

// ---------------------------------------------------------------------------
// Kernel 7: CrossMerge + LayerNorm(D_INNER) -> y_ln (B, L, D)
// grid (B, L), block 192 (thread = channel d)
// ---------------------------------------------------------------------------
__global__ __launch_bounds__(192) void k_merge_ln(
    const float* __restrict__ ysb, const float* __restrict__ lng,
    const float* __restrict__ lnb, float* __restrict__ yln)
{
    const int b = blockIdx.x, l = blockIdx.y, d = threadIdx.x;
    const int t = ((l & 63) << 6) | (l >> 6);
    float v = ysb[(((size_t)(b * 4 + 0)) * LL + l) * DI + d]
            + ysb[(((size_t)(b * 4 + 2)) * LL + (LL - 1 - l)) * DI + d]
            + ysb[(((size_t)(b * 4 + 1)) * LL + t) * DI + d]
            + ysb[(((size_t)(b * 4 + 3)) * LL + (LL - 1 - t)) * DI + d];

    __shared__ float sv[DI], sq[DI];
    __shared__ float smu, srstd;
    sv[d] = v; sq[d] = v * v;
    __syncthreads();
    if (d < 32) {
        float a = 0.f, q = 0.f;
        for (int i = d; i < DI; i += 32) { a += sv[i]; q += sq[i]; }
#pragma unroll
        for (int off = 16; off; off >>= 1) {
            a += __shfl_down(a, off);
            q += __shfl_down(q, off);
        }
        if (d == 0) {
            const float mu = a * (1.f / DI);
            smu = mu;
            srstd = rsqrtf(q * (1.f / DI) - mu * mu + 1e-5f);
        }
    }
    __syncthreads();
    yln[((size_t)b * LL + l) * DI + d] = (v - smu) * srstd * lng[d] + lnb[d];
}

// ---------------------------------------------------------------------------
// Kernel 8: out_proj (96x192) + BN -> d_out (B, 96, H, W) via WMMA
// grid (B, L/64), block 128
// ---------------------------------------------------------------------------
__global__ __launch_bounds__(128) void k_outproj(
    const float* __restrict__ yln, const float* __restrict__ ow,
    const float* __restrict__ bn_g, const float* __restrict__ bn_b,
    float* __restrict__ out)
{
    const int lane = threadIdx.x & 31, wv = threadIdx.x >> 5;
    const int b = blockIdx.x;
    const int l0 = blockIdx.y * 64 + wv * 16;
    const int n = lane & 15, hi = lane >> 4, kofs = hi * 2, ml = lane & 15;

    v8f acc[6] = {};
    const float* yb = yln + ((size_t)b * LL + l0 + n) * DI + kofs;
    for (int k0 = 0; k0 < DI; k0 += 4) {
        v2f bv = *(const v2f*)(yb + k0);
#pragma unroll
        for (int mt = 0; mt < 6; ++mt) {
            v2f av = *(const v2f*)(ow + (size_t)(mt * 16 + ml) * DI + k0 + kofs);
            acc[mt] = __builtin_amdgcn_wmma_f32_16x16x4_f32(
                false, av, false, bv, (short)0, acc[mt], false, false);
        }
    }
    const float inv = rsqrtf(1.f + 1e-5f);
#pragma unroll
    for (int mt = 0; mt < 6; ++mt)
#pragma unroll
        for (int r = 0; r < 8; ++r) {
            const int o = mt * 16 + r + hi * 8;
            out[(size_t)b * DM * LL + (size_t)o * LL + l0 + n] =
                acc[mt][r] * (bn_g[o] * inv) + bn_b[o];
        }
}

// ---------------------------------------------------------------------------
extern "C" void kernel_launch(void* const* d_in, const int* in_sizes, int n_in,
                              void* d_out, int out_size, void* d_ws, size_t ws_size,
                              hipStream_t stream)
{
    (void)in_sizes; (void)n_in; (void)out_size; (void)ws_size;
    const float* x      = (const float*)d_in[0];
    const float* in_w   = (const float*)d_in[1];
    const float* in_bng = (const float*)d_in[2];
    const float* in_bnb = (const float*)d_in[3];
    const float* dw_w   = (const float*)d_in[4];
    const float* dw_b   = (const float*)d_in[5];
    const float* xpw    = (const float*)d_in[6];
    const float* dtw    = (const float*)d_in[7];
    const float* dtb    = (const float*)d_in[8];
    const float* A_logs = (const float*)d_in[9];
    const float* Ds     = (const float*)d_in[10];
    const float* ln_g   = (const float*)d_in[11];
    const float* ln_b   = (const float*)d_in[12];
    const float* out_w  = (const float*)d_in[13];
    const float* o_bng  = (const float*)d_in[14];
    const float* o_bnb  = (const float*)d_in[15];
    float* out = (float*)d_out;

    // workspace layout (floats)
    float* ws = (float*)d_ws;
    const size_t sz_x1  = (size_t)BSZ * DI * LL;            // 1,572,864
    const size_t sz_u   = sz_x1;
    const size_t sz_dtr = (size_t)BSZ * KDIR * LL * DTR;    //   196,608
    const size_t sz_bc  = (size_t)BSZ * KDIR * LL * NS;     //   524,288
    const size_t sz_seg = (size_t)BSZ * KDIR * NSEG * DI * 32; // 3,145,728
    const size_t sz_ys  = (size_t)BSZ * KDIR * LL * DI;     // 6,291,456

    float* x1  = ws;                  ws += sz_x1;
    float* u   = ws;                  ws += sz_u;
    float* dtr = ws;                  ws += sz_dtr;
    float* Bsb = ws;                  ws += sz_bc;
    float* Csb = ws;                  ws += sz_bc;
    float* seg = ws;                  ws += sz_seg;
    float* ysb = ws;                  ws += sz_ys;
    float* yln = ws;                  /* sz_x1 */

    k_inproj <<<dim3(BSZ, LL / 64, 3), 128, 0, stream>>>(x, in_w, in_bng, in_bnb, x1);
    k_dwconv <<<(BSZ * DI * LL + 255) / 256, 256, 0, stream>>>(x1, dw_w, dw_b, u);
    k_xdbl   <<<dim3(BSZ * KDIR, LL / 64), 128, 0, stream>>>(u, xpw, dtr, Bsb, Csb);
    k_scan1  <<<dim3(BSZ * KDIR, NSEG), 192, 0, stream>>>(u, dtr, Bsb, dtw, dtb, A_logs, seg);
    k_scanmid<<<BSZ * KDIR, 192, 0, stream>>>(seg);
    k_scan2  <<<dim3(BSZ * KDIR, NSEG), 192, 0, stream>>>(u, dtr, Bsb, Csb, dtw, dtb,
                                                          A_logs, Ds, seg, ysb);
    k_merge_ln<<<dim3(BSZ, LL), 192, 0, stream>>>(ysb, ln_g, ln_b, yln);
    k_outproj<<<dim3(BSZ, LL / 64), 128, 0, stream>>>(yln, out_w, o_bng, o_bnb, out);
}